// FFTBias2d_8727373545480
// MI455X (gfx1250) — compile-verified
//
#include <hip/hip_runtime.h>

typedef __attribute__((ext_vector_type(2))) float v2f;
typedef __attribute__((ext_vector_type(8))) float v8f;

#define B_  16
#define S_  4096
#define H_  16
#define E_  64
#define SH  64    // s = sqrt(SEQ)
#define NW  127   // n = 2s-1

// ---------------------------------------------------------------------------
// Kernel 1: per (b,h) workgroup.
//  Phase 1: reduce v[b,:,h,:] (1 MB) into row sums u_s[e][y] and col sums
//           v_s[e][x] held in LDS (ds_add_f32 for the cross-thread sums).
//  Phase 2: Toeplitz conv as WMMA f32 16x16x4 matmuls:
//           Rx[e,k] = sum_m src[e,m] * w_h[(k-m+64) mod 127]
//           -> rxv_ws/rxu_ws laid out [b*H+h][k][e] (e innermost).
// ---------------------------------------------------------------------------
__global__ __launch_bounds__(256) void fft_bias_reduce_conv(
    const float* __restrict__ v, const float* __restrict__ w,
    float* __restrict__ rxv, float* __restrict__ rxu)
{
  __shared__ __align__(16) float vs[SH * SH];   // [e][x] col sums (A for RxV)
  __shared__ __align__(16) float us[SH * SH];   // [e][y] row sums (A for RxU)
  __shared__ __align__(16) float wl[128];       // w_h (127 used)

  const int bh  = blockIdx.x;
  const int b   = bh >> 4, h = bh & 15;
  const int tid = threadIdx.x;
  const int e   = tid & 63, q = tid >> 6;       // 4 q-groups cover x in 16s

  for (int i = tid; i < SH * SH; i += 256) us[i] = 0.0f;
  if (tid < NW) wl[tid] = w[h * NW + tid];
  __syncthreads();

  // ---- Phase 1: reductions over v[b, y*64+x, h, e] ----
  const float* vb = v + ((size_t)b * S_ * H_ + h) * (size_t)E_ + e;
  float accv[16];
  #pragma unroll
  for (int i = 0; i < 16; ++i) accv[i] = 0.0f;

  for (int y = 0; y < SH; ++y) {
    float su = 0.0f;
    const float* vt = vb + (size_t)(y * SH + q * 16) * (H_ * E_);
    #pragma unroll
    for (int i = 0; i < 16; ++i) {
      float val = vt[(size_t)i * (H_ * E_)];
      accv[i] += val;          // accumulate over y -> v_s[e][x]
      su      += val;          // accumulate over this thread's 16 x's
    }
    atomicAdd(&us[e * SH + y], su);   // ds_add_f32: u_s[e][y] over 4 q-groups
  }
  #pragma unroll
  for (int i = 0; i < 16; ++i) vs[e * SH + q * 16 + i] = accv[i];
  __syncthreads();

  // ---- Phase 2: WMMA f32 16x16x4 Toeplitz matmul ----
  const int wave = tid >> 5, lane = tid & 31;
  const int conv = wave >> 2;                 // 0: RxV (vs), 1: RxU (us)
  const int i0   = (wave & 3) * 16;           // output tile row block (e)
  const float* src  = conv ? us : vs;
  float*       outp = (conv ? rxu : rxv) + (size_t)bh * (SH * E_);

  const int m16     = lane & 15;              // M row (A) / N col (B) in frag
  const int kb      = (lane >> 4) << 1;       // K sub-offset: 0 or 2
  const int rowbase = i0 + ((lane >> 4) << 3);

  for (int jt = 0; jt < 4; ++jt) {
    const int j0 = jt * 16;                   // output tile col block (k)
    v8f acc = {0.f, 0.f, 0.f, 0.f, 0.f, 0.f, 0.f, 0.f};
    #pragma unroll
    for (int kk = 0; kk < 16; ++kk) {
      const int ka = kk * 4 + kb;             // K elements ka, ka+1
      // A-frag: A[i0+m16][ka], A[i0+m16][ka+1] (8B-aligned LDS pair)
      const float2 a2 = *reinterpret_cast<const float2*>(&src[(i0 + m16) * SH + ka]);
      // B-frag: B[ka][j0+n] = Wrel[(j0+n) - ka], Wrel[d] = w[(d+64) mod 127]
      int ix = j0 + m16 - ka + 64;            // range [2,127]
      float bx = wl[ix >= NW ? ix - NW : ix];
      float by = wl[ix - 1];                  // range [1,126], no wrap
      v2f af = {a2.x, a2.y};
      v2f bf = {bx, by};
      acc = __builtin_amdgcn_wmma_f32_16x16x4_f32(
          /*neg_a=*/false, af, /*neg_b=*/false, bf,
          /*c_mod=*/(short)0, acc, /*reuse_a=*/false, /*reuse_b=*/false);
    }
    // D layout: lane<16 -> rows i0+0..7, col j0+lane; lane>=16 -> rows i0+8..15
    const int col = j0 + (lane & 15);
    union { v8f v; float f[8]; } u; u.v = acc;
    #pragma unroll
    for (int r = 0; r < 8; ++r)
      outp[col * E_ + rowbase + r] = u.f[r];  // ws[bh][k=col][e=row]
  }
}

// ---------------------------------------------------------------------------
// Kernel 2: the bandwidth kernel. out1[b][t=y*64+x][h][e] =
//           rxv[bh][x][e] + rxu[bh][y][e].  float4 over e; rxv/rxu (8 MB)
//           stay L2-resident; ~268 MB streamed out.
// ---------------------------------------------------------------------------
__global__ __launch_bounds__(256) void fft_bias_broadcast(
    const float4* __restrict__ rxv, const float4* __restrict__ rxu,
    float4* __restrict__ out)
{
  const unsigned i  = blockIdx.x * 256u + threadIdx.x;  // < 16,777,216
  const unsigned e4 = i & 15u;
  const unsigned h  = (i >> 4) & 15u;
  const unsigned t  = (i >> 8) & 4095u;
  const unsigned b  = i >> 20;
  const unsigned x  = t & 63u, y = t >> 6;
  const unsigned base = (b * 16u + h) * 64u;
  float4 a = rxv[(base + x) * 16u + e4];
  float4 c = rxu[(base + y) * 16u + e4];
  float4 o;
  o.x = a.x + c.x; o.y = a.y + c.y; o.z = a.z + c.z; o.w = a.w + c.w;
  out[i] = o;
}

// ---------------------------------------------------------------------------
// Kernel 3: z_pb (tiny, 65,536 floats).
//   z[h][k] = 64 * sum_m o_[m] * w_h[(k-m+64) mod 127]
//   out2[t*16+h] = z[h][x] + z[h][y]
// ---------------------------------------------------------------------------
__global__ __launch_bounds__(256) void fft_bias_zpb(
    const float* __restrict__ w, const float* __restrict__ o_,
    float* __restrict__ out2)
{
  __shared__ float wl[H_ * NW];
  __shared__ float ol[SH];
  __shared__ float zz[H_ * SH];
  const int tid = threadIdx.x;
  for (int i = tid; i < H_ * NW; i += 256) wl[i] = w[i];
  if (tid < SH) ol[tid] = o_[tid];
  __syncthreads();

  for (int i = tid; i < H_ * SH; i += 256) {
    const int h = i >> 6, k = i & 63;
    float s = 0.f;
    #pragma unroll
    for (int m = 0; m < SH; ++m) {
      int j = k - m + 64;          // [k+1, k+64]
      if (j >= NW) j -= NW;
      s += ol[m] * wl[h * NW + j];
    }
    zz[i] = s * (float)SH;
  }
  __syncthreads();

  for (int i = tid; i < S_ * H_; i += 256) {
    const int t = i >> 4, h = i & 15;
    const int x = t & 63, y = t >> 6;
    out2[i] = zz[h * SH + x] + zz[h * SH + y];
  }
}

// ---------------------------------------------------------------------------
extern "C" void kernel_launch(void* const* d_in, const int* in_sizes, int n_in,
                              void* d_out, int out_size, void* d_ws, size_t ws_size,
                              hipStream_t stream)
{
  const float* v  = (const float*)d_in[0];   // (16, 4096, 16, 64) f32
  const float* w  = (const float*)d_in[1];   // (1, 16, 127) f32
  const float* o_ = (const float*)d_in[2];   // (64,) f32

  float* rxv  = (float*)d_ws;                       // [256][64][64] = 4 MB
  float* rxu  = rxv + (size_t)B_ * H_ * SH * E_;    // +4 MB
  float* out1 = (float*)d_out;                      // 16*4096*16*64
  float* out2 = out1 + (size_t)B_ * S_ * H_ * E_;   // 4096*16

  fft_bias_reduce_conv<<<B_ * H_, 256, 0, stream>>>(v, w, rxv, rxu);
  fft_bias_broadcast<<<(B_ * S_ * H_ * E_ / 4) / 256, 256, 0, stream>>>(
      (const float4*)rxv, (const float4*)rxu, (float4*)out1);
  fft_bias_zpb<<<1, 256, 0, stream>>>(w, o_, out2);
}